// MyLoss_83854941487213
// MI455X (gfx1250) — compile-verified
//
#include <hip/hip_runtime.h>
#include <hip/hip_bf16.h>
#include <math.h>

// ---------------------------------------------------------------------------
// YOLO-style loss for MI455X (gfx1250, wave32).
//  - Dense term (lobj) reads ONLY channels 0..3 of each scale (2.15 MB total),
//    as float4 non-temporal loads (global_load_b128, TH=NT: read-once data).
//  - Sparse terms (lbox/lcls) gather 4+80 channels at 32 GT cells per batch.
//  - Cross-lane reduction uses V_WMMA_F32_16X16X4_F32: A holds per-lane
//    partials, B = ones(4x16)  =>  D[m][n] = partial[m] + partial[m+16].
//    Summing the 8 D VGPRs per lane then one shfl_xor(16) gives the wave sum.
//  - sigmoid uses v_exp_f32 + v_rcp_f32 (~1 ulp; negligible vs. reduction).
// ---------------------------------------------------------------------------

typedef __attribute__((ext_vector_type(2))) float v2f;
typedef __attribute__((ext_vector_type(4))) float v4f;
typedef __attribute__((ext_vector_type(8))) float v8f;

__device__ __forceinline__ float wave_sum_wmma(float x) {
    // A (16x4 f32): lane L<16 -> row M=L, {K0,K1}; lane L>=16 -> row M=L-16, {K2,K3}
    v2f a; a[0] = x; a[1] = 0.0f;          // A[m][0]=partial[m], A[m][2]=partial[m+16]
    v2f b; b[0] = 1.0f; b[1] = 1.0f;       // B = ones(4x16) (layout-independent)
    v8f c = {0.f, 0.f, 0.f, 0.f, 0.f, 0.f, 0.f, 0.f};
    // 8 args: (neg_a, A, neg_b, B, c_mod, C, reuse_a, reuse_b)
    c = __builtin_amdgcn_wmma_f32_16x16x4_f32(false, a, false, b, (short)0, c,
                                              false, false);
    // D[m][n] = rowsum_m; lanes 0..15 see M = vgpr, lanes 16..31 see M = vgpr+8.
    float s = c[0] + c[1] + c[2] + c[3] + c[4] + c[5] + c[6] + c[7];
    // lanes 0..15 hold sum(rows 0..7), lanes 16..31 hold sum(rows 8..15)
    s += __shfl_xor(s, 16, 32);
    return s;   // full 32-lane sum in every lane
}

__device__ __forceinline__ float block_sum(float x, float* wsum /*>=8 floats*/) {
    float w = wave_sum_wmma(x);
    const int lane = threadIdx.x & 31;
    const int wid  = threadIdx.x >> 5;
    if (lane == 0) wsum[wid] = w;
    __syncthreads();
    float r = 0.f;
    if (threadIdx.x == 0) {
        const int nw = blockDim.x >> 5;
        for (int i = 0; i < nw; ++i) r += wsum[i];
    }
    return r;   // valid in thread 0 only
}

__global__ void yolo_init_kernel(float* __restrict__ out) {
    out[0] = 0.0f;
}

// lobj: (1/(bs*H*W)) * sum_cells (sigmoid(mean(ch0..3)) - is_gt)^2, pre-weighted.
// Each thread owns 4 consecutive cells (one float4 per channel plane).
__global__ void yolo_obj_kernel(const float* __restrict__ pred,
                                const float* __restrict__ gtb,
                                int H, int W, int nchunks, float wobj,
                                float* __restrict__ out) {
    const int b   = blockIdx.x;
    const int HW  = H * W;
    const int HWv = HW >> 2;               // plane stride in float4 units

    __shared__ int   gtcell[32];
    __shared__ float wsum[8];

    if (threadIdx.x < 32) {
        const float* g = gtb + ((size_t)b * 32 + threadIdx.x) * 4;
        float cx = g[0] * (float)W;
        float cy = g[1] * (float)H;
        int gx = (int)cx; gx = min(max(gx, 0), W - 1);
        int gy = (int)cy; gy = min(max(gy, 0), H - 1);
        gtcell[threadIdx.x] = gy * W + gx;
    }
    __syncthreads();

    const int cpb   = (HWv + nchunks - 1) / nchunks;
    const int start = blockIdx.y * cpb;
    const int end   = min(HWv, start + cpb);
    const v4f* c0 = (const v4f*)(pred + (size_t)b * 144 * (size_t)HW);

    float partial = 0.f;
    for (int v = start + (int)threadIdx.x; v < end; v += blockDim.x) {
        // speculative prefetch of next iteration (emits global_prefetch_b8)
        __builtin_prefetch(c0 + v + blockDim.x, 0, 1);
        // read-once streaming data: non-temporal global_load_b128 x4
        v4f a0 = __builtin_nontemporal_load(c0 + v);
        v4f a1 = __builtin_nontemporal_load(c0 + v + HWv);
        v4f a2 = __builtin_nontemporal_load(c0 + v + 2 * HWv);
        v4f a3 = __builtin_nontemporal_load(c0 + v + 3 * HWv);

        const int i = v << 2;              // base cell index of this float4
        unsigned mask = 0u;                // bit j set <=> cell i+j is a GT cell
#pragma unroll
        for (int o = 0; o < 32; ++o) {
            unsigned d = (unsigned)(gtcell[o] - i);
            mask |= (d < 4u) ? (1u << (d & 3u)) : 0u;
        }
#pragma unroll
        for (int j = 0; j < 4; ++j) {
            float m = 0.25f * (a0[j] + a1[j] + a2[j] + a3[j]);
            float s = __builtin_amdgcn_rcpf(1.0f + __expf(-m));  // v_rcp/v_exp
            float t = (float)((mask >> j) & 1u);
            float d = s - t;
            partial += d * d;
        }
    }

    float bsum = block_sum(partial, wsum);
    if (threadIdx.x == 0) atomicAdd(out, wobj * bsum);
}

// lbox + lcls at the 32 GT cells of one batch. 8 threads per object:
// sub 0 also does the 4 box channels; every sub does 10 class channels.
__global__ void yolo_gather_kernel(const float* __restrict__ pred,
                                   const float* __restrict__ gtb,
                                   const int* __restrict__ gtc,
                                   int H, int W, float wbox, float wcls,
                                   float* __restrict__ out) {
    const int b  = blockIdx.x;
    const int HW = H * W;

    __shared__ int   cell_s[32];
    __shared__ float boxt[32][4];
    __shared__ int   cls_s[32];
    __shared__ float wsum[8];

    if (threadIdx.x < 32) {
        const int o = threadIdx.x;
        const float* g = gtb + ((size_t)b * 32 + o) * 4;
        float cx = g[0] * (float)W;
        float cy = g[1] * (float)H;
        int gx = (int)cx; gx = min(max(gx, 0), W - 1);
        int gy = (int)cy; gy = min(max(gy, 0), H - 1);
        cell_s[o]  = gy * W + gx;
        boxt[o][0] = cx - (float)gx;       // cx - gx
        boxt[o][1] = cy - (float)gy;       // cy - gy
        boxt[o][2] = g[2];                 // (gw*W)/W = raw w
        boxt[o][3] = g[3];                 // (gh*H)/H = raw h
        cls_s[o]   = gtc[b * 32 + o];
    }
    __syncthreads();

    const int o   = threadIdx.x >> 3;      // 0..31
    const int sub = threadIdx.x & 7;       // 0..7
    const float* base = pred + (size_t)b * 144 * (size_t)HW + cell_s[o];

    float partial = 0.f;
    if (sub == 0) {                         // box term: channels 0..3, mean of sq
        float sb = 0.f;
#pragma unroll
        for (int j = 0; j < 4; ++j) {
            float d = base[(size_t)j * HW] - boxt[o][j];
            sb += d * d;
        }
        partial += wbox * 0.25f * sb;
    }
    {                                       // cls term: channels 64..143
        const int cls = cls_s[o];
        float sc = 0.f;
#pragma unroll
        for (int kk = 0; kk < 10; ++kk) {
            int   k = sub * 10 + kk;
            float c = base[(size_t)(64 + k) * HW];
            // numerically stable softplus (precise path: only 123k evals)
            float sp = fmaxf(c, 0.f) + log1pf(expf(-fabsf(c)));
            sc += sp - ((cls == k) ? c : 0.f);
        }
        partial += wcls * sc;               // wcls already has 1/80 folded in
    }

    float bsum = block_sum(partial, wsum);
    if (threadIdx.x == 0) atomicAdd(out, bsum);
}

extern "C" void kernel_launch(void* const* d_in, const int* in_sizes, int n_in,
                              void* d_out, int out_size, void* d_ws, size_t ws_size,
                              hipStream_t stream) {
    const float* p0  = (const float*)d_in[0];
    const float* p1  = (const float*)d_in[1];
    const float* p2  = (const float*)d_in[2];
    const float* gtb = (const float*)d_in[3];
    const int*   gtc = (const int*)d_in[4];
    float* out = (float*)d_out;

    const int bs = in_sizes[3] / (32 * 4);   // gt_boxes is (bs, 32, 4)

    yolo_init_kernel<<<1, 1, 0, stream>>>(out);

    const float* preds[3] = {p0, p1, p2};
    const int    HS[3]    = {80, 40, 20};
    const int    WS[3]    = {80, 40, 20};

    const float LB = 7.5f, LO = 1.0f, LC = 0.5f;
    const float wbox = LB / 3.0f;
    const float wcls = (LC / 3.0f) * (1.0f / 80.0f);

    for (int s = 0; s < 3; ++s) {
        const int H = HS[s], W = WS[s], HW = H * W;
        const int HWv = HW >> 2;
        const int nchunks = (HWv + 511) / 512;   // ~512 float4 cells per block
        const float wobj = (LO / 3.0f) / ((float)bs * (float)HW);

        dim3 grid(bs, nchunks);
        yolo_obj_kernel<<<grid, 256, 0, stream>>>(preds[s], gtb, H, W, nchunks,
                                                  wobj, out);
        yolo_gather_kernel<<<bs, 256, 0, stream>>>(preds[s], gtb, gtc, H, W,
                                                   wbox, wcls, out);
    }
}